// mincut_net_54726473286129
// MI455X (gfx1250) — compile-verified
//
#include <hip/hip_runtime.h>

typedef float v2f __attribute__((ext_vector_type(2)));
typedef float v8f __attribute__((ext_vector_type(8)));

#define N_NODES 100000
#define N_EDGES 1600000
#define HID     32
#define KCL     4
#define OUTC    32

// ---------------------------------------------------------------------------
// 1. init: agg = 0, deg = 1 (self loop), sx accumulator = 0
// ---------------------------------------------------------------------------
__global__ void init_kernel(float* __restrict__ agg, float* __restrict__ deg,
                            float* __restrict__ sx) {
    int tid = blockIdx.x * blockDim.x + threadIdx.x;
    if (tid < N_NODES * HID) agg[tid] = 0.0f;
    if (tid < N_NODES)       deg[tid] = 1.0f;
    if (tid < KCL * HID)     sx[tid]  = 0.0f;
}

// ---------------------------------------------------------------------------
// 2. x = emb[node_types]
// ---------------------------------------------------------------------------
__global__ void embed_kernel(const int* __restrict__ nt,
                             const float* __restrict__ emb,
                             float* __restrict__ x) {
    int tid = blockIdx.x * blockDim.x + threadIdx.x;
    if (tid >= N_NODES * HID) return;
    int n = tid >> 5, j = tid & 31;
    x[tid] = emb[nt[n] * HID + j];
}

// ---------------------------------------------------------------------------
// 3. deg[dst] += 1 per edge
// ---------------------------------------------------------------------------
__global__ void deg_kernel(const int* __restrict__ ei, float* __restrict__ deg) {
    int e = blockIdx.x * blockDim.x + threadIdx.x;
    if (e >= N_EDGES) return;
    atomicAdd(&deg[ei[N_EDGES + e]], 1.0f);
}

// ---------------------------------------------------------------------------
// 4. dinv = rsqrt(deg) in place
// ---------------------------------------------------------------------------
__global__ void rsqrt_kernel(float* __restrict__ deg) {
    int n = blockIdx.x * blockDim.x + threadIdx.x;
    if (n < N_NODES) deg[n] = rsqrtf(deg[n]);
}

// ---------------------------------------------------------------------------
// 5. xw = x @ W_gcn   (fp32 WMMA 16x16x4, one 16-row tile per wave, 2 N-tiles)
//    A 16x4 f32 layout: lane M = lane&15, lanes16-31 hold K+2 (VGPR0=K, VGPR1=K+1)
//    B 4x16 layout:     lane N = lane&15, lanes16-31 hold K+2
//    C/D:               VGPR r holds M = r (+8 for lanes16-31), N = lane&15
// ---------------------------------------------------------------------------
__global__ void gemm_xw_kernel(const float* __restrict__ x,
                               const float* __restrict__ Wg,
                               float* __restrict__ xw) {
    __shared__ float Wl[HID * HID];
    int tid = threadIdx.x;
    for (int i = tid; i < HID * HID; i += blockDim.x) Wl[i] = Wg[i];
    __syncthreads();

    const int lane = tid & 31;
    const int tile = blockIdx.x * (blockDim.x >> 5) + (tid >> 5);
    const int row0 = tile * 16;
    if (row0 >= N_NODES) return;      // wave-uniform: EXEC stays all-ones

    const int m   = lane & 15;
    const int kh  = (lane < 16) ? 0 : 2;
    const int row = row0 + m;

    v8f c0 = {}; v8f c1 = {};
    for (int kk = 0; kk < HID; kk += 4) {
        int kb = kk + kh;
        v2f a;  a.x  = x[row * HID + kb];     a.y  = x[row * HID + kb + 1];
        v2f b0; b0.x = Wl[kb * HID + m];      b0.y = Wl[(kb + 1) * HID + m];
        v2f b1; b1.x = Wl[kb * HID + m + 16]; b1.y = Wl[(kb + 1) * HID + m + 16];
        c0 = __builtin_amdgcn_wmma_f32_16x16x4_f32(false, a, false, b0, (short)0, c0, false, false);
        c1 = __builtin_amdgcn_wmma_f32_16x16x4_f32(false, a, false, b1, (short)0, c1, false, false);
    }
    const int rb = (lane < 16) ? 0 : 8;
    for (int r = 0; r < 8; ++r) {
        int ro = row0 + rb + r;
        xw[ro * HID + m]      = c0[r];
        xw[ro * HID + m + 16] = c1[r];
    }
}

// ---------------------------------------------------------------------------
// 6. agg[dst] += xw[src] * dinv[src]*dinv[dst]   (lane per (edge, feature))
// ---------------------------------------------------------------------------
__global__ void edge_agg_kernel(const int* __restrict__ ei,
                                const float* __restrict__ xw,
                                const float* __restrict__ dinv,
                                float* __restrict__ agg) {
    long long tid = (long long)blockIdx.x * blockDim.x + threadIdx.x;
    if (tid >= (long long)N_EDGES * HID) return;
    int e = (int)(tid >> 5);
    int j = (int)(tid & 31);
    int s = ei[e];
    int d = ei[N_EDGES + e];
    float nrm = dinv[s] * dinv[d];
    atomicAdd(&agg[d * HID + j], xw[s * HID + j] * nrm);
}

// ---------------------------------------------------------------------------
// 7. per-node: xg = relu(agg + xw*dinv^2 + b); s = xg@W_pool + b_pool;
//    softmax(K=4), argmax -> idx; sx[k][h] += soft_k * xg[h] (LDS staged)
// ---------------------------------------------------------------------------
__global__ void node_post_kernel(const float* __restrict__ agg,
                                 const float* __restrict__ xw,
                                 const float* __restrict__ dinv,
                                 const float* __restrict__ b_gcn,
                                 const float* __restrict__ W_pool,
                                 const float* __restrict__ b_pool,
                                 float* __restrict__ xg,
                                 int*   __restrict__ idx,
                                 float* __restrict__ sx) {
    __shared__ float sxl[KCL * HID];
    int tid = threadIdx.x;
    if (tid < KCL * HID) sxl[tid] = 0.0f;
    __syncthreads();

    const int lane  = tid & 31;
    const int wpb   = blockDim.x >> 5;
    const int wave  = blockIdx.x * wpb + (tid >> 5);
    const int nwave = gridDim.x * wpb;

    const float wp0 = W_pool[lane * KCL + 0];
    const float wp1 = W_pool[lane * KCL + 1];
    const float wp2 = W_pool[lane * KCL + 2];
    const float wp3 = W_pool[lane * KCL + 3];
    const float bp0 = b_pool[0], bp1 = b_pool[1], bp2 = b_pool[2], bp3 = b_pool[3];
    const float bg  = b_gcn[lane];

    for (int n = wave; n < N_NODES; n += nwave) {
        float di = dinv[n];
        float v  = agg[n * HID + lane] + xw[n * HID + lane] * di * di + bg;
        v = fmaxf(v, 0.0f);
        xg[n * HID + lane] = v;

        float s0 = v * wp0, s1 = v * wp1, s2 = v * wp2, s3 = v * wp3;
        for (int off = 16; off > 0; off >>= 1) {
            s0 += __shfl_xor(s0, off, 32);
            s1 += __shfl_xor(s1, off, 32);
            s2 += __shfl_xor(s2, off, 32);
            s3 += __shfl_xor(s3, off, 32);
        }
        s0 += bp0; s1 += bp1; s2 += bp2; s3 += bp3;

        float mx = fmaxf(fmaxf(s0, s1), fmaxf(s2, s3));
        float e0 = expf(s0 - mx), e1 = expf(s1 - mx);
        float e2 = expf(s2 - mx), e3 = expf(s3 - mx);
        float inv = 1.0f / (e0 + e1 + e2 + e3);
        e0 *= inv; e1 *= inv; e2 *= inv; e3 *= inv;

        int am = 0; float bv = s0;                 // first-max semantics
        if (s1 > bv) { bv = s1; am = 1; }
        if (s2 > bv) { bv = s2; am = 2; }
        if (s3 > bv) { bv = s3; am = 3; }
        if (lane == 0) idx[n] = am;

        atomicAdd(&sxl[0 * HID + lane], e0 * v);
        atomicAdd(&sxl[1 * HID + lane], e1 * v);
        atomicAdd(&sxl[2 * HID + lane], e2 * v);
        atomicAdd(&sxl[3 * HID + lane], e3 * v);
    }
    __syncthreads();
    if (tid < KCL * HID) atomicAdd(&sx[tid], sxl[tid]);
}

// ---------------------------------------------------------------------------
// 8. cfeat = (relu(sx @ W1 + b1)) @ W2 + b2   [4,32] — single block
// ---------------------------------------------------------------------------
__global__ void cluster_mlp_kernel(const float* __restrict__ sx,
                                   const float* __restrict__ W1, const float* __restrict__ b1,
                                   const float* __restrict__ W2, const float* __restrict__ b2,
                                   float* __restrict__ cfeat) {
    __shared__ float sxl[KCL * HID];
    __shared__ float hid[KCL * HID];
    int tid = threadIdx.x;                 // 128 threads: k = tid>>5, h = tid&31
    sxl[tid] = sx[tid];
    __syncthreads();
    int k = tid >> 5, h = tid & 31;
    float acc = b1[h];
    for (int j = 0; j < HID; ++j) acc += sxl[k * HID + j] * W1[j * HID + h];
    hid[tid] = fmaxf(acc, 0.0f);
    __syncthreads();
    float acc2 = b2[h];
    for (int j = 0; j < HID; ++j) acc2 += hid[k * HID + j] * W2[j * OUTC + h];
    cfeat[tid] = acc2;
}

// ---------------------------------------------------------------------------
// 9. out = [xg | cfeat[idx]] @ W3 + b3   (fp32 WMMA, K=64)
// ---------------------------------------------------------------------------
__global__ void gemm_out_kernel(const float* __restrict__ xg,
                                const int*   __restrict__ idx,
                                const float* __restrict__ cfeat,
                                const float* __restrict__ W3,
                                const float* __restrict__ b3,
                                float* __restrict__ out) {
    __shared__ float W3l[(HID + OUTC) * OUTC];
    __shared__ float cfl[KCL * HID];
    __shared__ float b3l[OUTC];
    int tid = threadIdx.x;
    for (int i = tid; i < (HID + OUTC) * OUTC; i += blockDim.x) W3l[i] = W3[i];
    if (tid < KCL * HID) cfl[tid] = cfeat[tid];
    if (tid < OUTC)      b3l[tid] = b3[tid];
    __syncthreads();

    const int lane = tid & 31;
    const int tile = blockIdx.x * (blockDim.x >> 5) + (tid >> 5);
    const int row0 = tile * 16;
    if (row0 >= N_NODES) return;      // wave-uniform

    const int m   = lane & 15;
    const int kh  = (lane < 16) ? 0 : 2;
    const int row = row0 + m;
    const int ci  = idx[row];

    v8f c0 = {}; v8f c1 = {};
    for (int kk = 0; kk < HID + OUTC; kk += 4) {
        int kb = kk + kh;           // kb even => kb,kb+1 stay in same half
        v2f a;
        if (kb < HID) { a.x = xg[row * HID + kb];        a.y = xg[row * HID + kb + 1]; }
        else          { a.x = cfl[ci * HID + kb - HID];  a.y = cfl[ci * HID + kb - HID + 1]; }
        v2f b0; b0.x = W3l[kb * OUTC + m];      b0.y = W3l[(kb + 1) * OUTC + m];
        v2f b1; b1.x = W3l[kb * OUTC + m + 16]; b1.y = W3l[(kb + 1) * OUTC + m + 16];
        c0 = __builtin_amdgcn_wmma_f32_16x16x4_f32(false, a, false, b0, (short)0, c0, false, false);
        c1 = __builtin_amdgcn_wmma_f32_16x16x4_f32(false, a, false, b1, (short)0, c1, false, false);
    }
    const int rb = (lane < 16) ? 0 : 8;
    for (int r = 0; r < 8; ++r) {
        int ro = row0 + rb + r;
        out[ro * OUTC + m]      = c0[r] + b3l[m];
        out[ro * OUTC + m + 16] = c1[r] + b3l[m + 16];
    }
}

// ---------------------------------------------------------------------------
extern "C" void kernel_launch(void* const* d_in, const int* in_sizes, int n_in,
                              void* d_out, int out_size, void* d_ws, size_t ws_size,
                              hipStream_t stream) {
    const int*   node_types = (const int*)d_in[0];
    const int*   edge_index = (const int*)d_in[1];
    const float* emb    = (const float*)d_in[2];
    const float* W_gcn  = (const float*)d_in[3];
    const float* b_gcn  = (const float*)d_in[4];
    const float* W_pool = (const float*)d_in[5];
    const float* b_pool = (const float*)d_in[6];
    const float* W1 = (const float*)d_in[7];
    const float* b1 = (const float*)d_in[8];
    const float* W2 = (const float*)d_in[9];
    const float* b2 = (const float*)d_in[10];
    const float* W3 = (const float*)d_in[11];
    const float* b3 = (const float*)d_in[12];
    float* out = (float*)d_out;

    // workspace layout (floats): ~39 MB total
    float* ws   = (float*)d_ws;
    float* xbuf = ws;                                  // [N,32] emb x, later xg
    float* xw   = ws + (size_t)N_NODES * HID;          // [N,32]
    float* agg  = ws + (size_t)2 * N_NODES * HID;      // [N,32]
    float* dinv = ws + (size_t)3 * N_NODES * HID;      // [N] deg -> dinv
    int*   idx  = (int*)(dinv + N_NODES);              // [N]
    float* sx   = dinv + 2 * (size_t)N_NODES;          // [4,32]
    float* cfeat= sx + KCL * HID;                      // [4,32]

    const int T = 256;
    const int gElem  = (N_NODES * HID + T - 1) / T;
    const int gEdge  = (N_EDGES + T - 1) / T;
    const int gNode  = (N_NODES + T - 1) / T;
    const int nTiles = N_NODES / 16;                   // N divisible by 16
    const int gGemm  = (nTiles + (T / 32) - 1) / (T / 32);
    const long long eWork = (long long)N_EDGES * HID;
    const int gEAgg  = (int)((eWork + T - 1) / T);

    init_kernel<<<gElem, T, 0, stream>>>(agg, dinv, sx);
    embed_kernel<<<gElem, T, 0, stream>>>(node_types, emb, xbuf);
    deg_kernel<<<gEdge, T, 0, stream>>>(edge_index, dinv);
    rsqrt_kernel<<<gNode, T, 0, stream>>>(dinv);
    gemm_xw_kernel<<<gGemm, T, 0, stream>>>(xbuf, W_gcn, xw);
    edge_agg_kernel<<<gEAgg, T, 0, stream>>>(edge_index, xw, dinv, agg);
    node_post_kernel<<<512, T, 0, stream>>>(agg, xw, dinv, b_gcn, W_pool, b_pool,
                                            xbuf, idx, sx);
    cluster_mlp_kernel<<<1, KCL * HID, 0, stream>>>(sx, W1, b1, W2, b2, cfeat);
    gemm_out_kernel<<<gGemm, T, 0, stream>>>(xbuf, idx, cfeat, W3, b3, out);
}